// CTCLoss2D_16338055594153
// MI455X (gfx1250) — compile-verified
//
#include <hip/hip_runtime.h>
#include <cstdint>

// Problem constants (fixed by the reference)
#define TT 128
#define HH 16
#define NN 64
#define CC 256
#define SS 129          // 2*64+1 expanded target length
#define NTHREADS 160    // 5 wave32s; threads 0..128 own lattice position s

#define LOG_TINY_F  (-87.3365447505531f)   // log(float32 tiny)
#define TINY_F      (1.1754943508222875e-38f)
#define LOG_H_F     (2.772588722239781f)   // log(16)

// log_add(a,b) = max + log1p(exp(min-max)); fast trans ops (v_exp_f32/v_log_f32)
__device__ __forceinline__ float log_add_f(float a, float b) {
    float x = fmaxf(a, b);
    float y = fminf(a, b);
    return x + __logf(1.0f + __expf(y - x));
}

// CDNA5 async global->LDS copy, 16B per lane, non-temporal (stream-once data).
// VDST VGPR = LDS byte address, VADDR pair = 64-bit global address. ASYNCcnt.
__device__ __forceinline__ void async_load_b128(uint32_t lds_off, const void* gptr) {
    asm volatile("global_load_async_to_lds_b128 %0, %1, off th:TH_LOAD_NT"
                 :: "v"(lds_off), "v"((unsigned long long)(uintptr_t)gptr)
                 : "memory");
}

__device__ __forceinline__ void wait_async0() {
    asm volatile("s_wait_asynccnt 0" ::: "memory");
}

// Low 32 bits of a generic pointer to __shared__ data == LDS byte offset
__device__ __forceinline__ uint32_t lds_addr(const void* p) {
    return (uint32_t)(uintptr_t)p;
}

__global__ __launch_bounds__(NTHREADS)
void ctc2d_kernel(const float* __restrict__ mask,      // [T,H,N] log-softmax over H
                  const float* __restrict__ classify,  // [T,H,N,C] log-softmax over C
                  const int*   __restrict__ targets,   // [N,64]
                  const int*   __restrict__ tlen,      // [N]
                  float*       __restrict__ out)       // [N]
{
    const int n   = blockIdx.x;
    const int tid = threadIdx.x;
    const int s   = tid;
    const bool active = (s < SS);

    __shared__ float cls_buf[2][HH][CC];   // double-buffered classify[t] rows (32 KB)
    __shared__ float maskv[2][HH];         // double-buffered mask[t] row
    __shared__ float hs_lds[SS + 3];
    __shared__ int   cc_lds[SS];           // reachability frontier cumsum

    // Per-thread expanded-target values (pure function of targets; no LDS needed)
    int et_s = 0;
    bool skip_s = false;
    if (active) {
        et_s = (s & 1) ? targets[n * 64 + (s >> 1)] : 0;
        if (s >= 2) {
            int et_m2 = (s & 1) ? targets[n * 64 + ((s - 2) >> 1)] : 0;
            skip_s = (et_s != et_m2);
        }
    }

    // Prefetch classify[1] rows -> cls_buf[1] ASAP (16 rows * 1KB = 16KB)
    for (int j = tid; j < HH * (CC / 4); j += NTHREADS) {
        const int h  = j >> 6;        // 64 float4 per row
        const int c4 = j & 63;
        const void* g = (const void*)(classify + (((size_t)1 * HH + h) * NN + n) * CC + (size_t)c4 * 4);
        async_load_b128(lds_addr(&cls_buf[1][h][c4 * 4]), g);
    }

    // Thread 0: cc[j] prefix sum. f[0]=f[1]=msf0+1 with msf0=(et2!=et0)=0 (both blank).
    if (tid == 0) {
        const int base = n * 64;
        int cc = 1;          // cc[0]
        cc_lds[0] = cc;
        cc += 1;             // cc[1]
        cc_lds[1] = cc;
        for (int j = 2; j < SS; ++j) {
            int d = 1;
            if (j & 1) {     // only label positions can differ across the skip edge
                int a = targets[base + (j >> 1)];
                int b = targets[base + ((j - 2) >> 1)];
                d = (a != b) ? 2 : 1;
            }
            cc += d;
            cc_lds[j] = cc;
        }
    }

    // mask[0] row
    if (tid < HH) maskv[0][tid] = mask[(size_t)tid * NN + n];

    // p0: s<2 gathered from classify[0]; s>=2 = LOG_TINY - log(H)
    float prob[HH];
    if (active) {
        if (s < 2) {
            #pragma unroll
            for (int h = 0; h < HH; ++h)
                prob[h] = classify[(((size_t)h) * NN + n) * CC + et_s];
        } else {
            #pragma unroll
            for (int h = 0; h < HH; ++h)
                prob[h] = LOG_TINY_F - LOG_H_F;
        }
    }

    // Scan: step i computes alpha at time t=i+1 using mask[i], classify[i+1].
    // Async pipeline: asyncs issued at top of step i fill cls_buf[i&1] with
    // classify[i+2]; they are waited at the top of step i+1, whose gather reads
    // that slot. Overlap window = one full step of compute.
    for (int i = 0; i < TT - 1; ++i) {
        const int cur = (i + 1) & 1;   // cls_buf slot holding classify[i+1]
        const int mv  = i & 1;         // maskv slot holding mask[i]

        wait_async0();                 // asyncs issued at top of step i-1 landed
        __syncthreads();               // visible to all; prior readers of slot i&1 done

        // Issue prefetch classify[i+2] -> cls_buf[i&1] (consumed next iteration)
        if (i + 2 <= TT - 1) {
            for (int j = tid; j < HH * (CC / 4); j += NTHREADS) {
                const int h  = j >> 6;
                const int c4 = j & 63;
                const void* g = (const void*)(classify +
                    (((size_t)(i + 2) * HH + h) * NN + n) * CC + (size_t)c4 * 4);
                async_load_b128(lds_addr(&cls_buf[i & 1][h][c4 * 4]), g);
            }
        }
        // Prefetch mask[i+1] row
        if (tid < HH)
            maskv[(i + 1) & 1][tid] = mask[((size_t)(i + 1) * HH + tid) * NN + n];

        // hs[s] = log(max(sum_h exp(prob+mask), TINY)), masked by reachability
        float hs = LOG_TINY_F;
        if (active) {
            float sum = 0.0f;
            #pragma unroll
            for (int h = 0; h < HH; ++h)
                sum += __expf(prob[h] + maskv[mv][h]);
            hs = __logf(fmaxf(sum, TINY_F));
            if (s > cc_lds[i + 1]) hs = LOG_TINY_F;
            hs_lds[s] = hs;
        }
        __syncthreads();

        // CTC recurrence over neighbors
        float newp = 0.0f;
        if (active) {
            if (s == 0) {
                newp = hs;
            } else {
                float v = log_add_f(hs, hs_lds[s - 1]);
                if (skip_s) v = log_add_f(v, hs_lds[s - 2]);
                newp = v;
            }
        }

        // Gather from LDS-staged classify[i+1]
        if (active) {
            #pragma unroll
            for (int h = 0; h < HH; ++h)
                prob[h] = newp + cls_buf[cur][h][et_s];
        }
    }

    wait_async0();
    __syncthreads();

    // Final: res[s] = iterated log_add over h of (prob + mask[T-1])
    if (active) {
        const int mv = (TT - 1) & 1;
        float res = prob[0] + maskv[mv][0];
        #pragma unroll
        for (int h = 1; h < HH; ++h)
            res = log_add_f(res, prob[h] + maskv[mv][h]);
        hs_lds[s] = res;
    }
    __syncthreads();

    if (tid == 0) {
        const int L  = tlen[n];
        const int Ls = 2 * L + 1;
        const float g1 = hs_lds[Ls - 1];
        const float g2 = hs_lds[Ls - 2];
        out[n] = -(log_add_f(g1, g2) / (float)L);
    }
}

extern "C" void kernel_launch(void* const* d_in, const int* in_sizes, int n_in,
                              void* d_out, int out_size, void* d_ws, size_t ws_size,
                              hipStream_t stream) {
    const float* mask     = (const float*)d_in[0];   // T*H*N
    const float* classify = (const float*)d_in[1];   // T*H*N*C
    const int*   targets  = (const int*)d_in[2];     // N*64
    // d_in[3] = input_lengths (all == T, unused)
    const int*   tlen     = (const int*)d_in[4];     // N
    float* out = (float*)d_out;

    ctc2d_kernel<<<NN, NTHREADS, 0, stream>>>(mask, classify, targets, tlen, out);
}